// ProteinDockingModel_73658689126892
// MI455X (gfx1250) — compile-verified
//
#include <hip/hip_runtime.h>
#include <hip/hip_bf16.h>

// ---------------------------------------------------------------------------
// Problem constants (from the reference)
// ---------------------------------------------------------------------------
#define NN   50000      // nodes (multiple of 16: 3125*16)
#define EE   800000     // edges (multiple of 8)
#define FF   26         // input features
#define HH   128        // hidden dim
#define LNEPS 1e-5f

typedef __attribute__((ext_vector_type(2))) float v2f;
typedef __attribute__((ext_vector_type(8))) float v8f;

// ---------------------------------------------------------------------------
// Input projection: X = relu(Xf @ Wi + bi)   [N,26] @ [26,128]
// one row per block, 128 threads
// ---------------------------------------------------------------------------
__global__ void __launch_bounds__(128)
k_input_proj(const float* __restrict__ Xf, const float* __restrict__ Wi,
             const float* __restrict__ bi, float* __restrict__ X)
{
    __shared__ float xr[FF];
    const int row = blockIdx.x;
    const int j   = threadIdx.x;
    if (j < FF) xr[j] = Xf[row * FF + j];
    __syncthreads();
    float acc = bi[j];
#pragma unroll
    for (int k = 0; k < FF; ++k) acc = fmaf(xr[k], Wi[k * HH + j], acc);
    X[row * HH + j] = fmaxf(acc, 0.f);
}

// ---------------------------------------------------------------------------
// fp32 WMMA GEMM: xt = X @ W + b ; agg = xt   (self term of the conv)
// [N,128] @ [128,128].  Block = 256 threads = 8 waves; block owns 16 rows,
// wave w owns output columns [16w, 16w+16).  K-chain of 32 x V_WMMA_F32_16X16X4_F32.
// ---------------------------------------------------------------------------
__global__ void __launch_bounds__(256)
k_gemm_wmma_f32(const float* __restrict__ X, const float* __restrict__ W,
                const float* __restrict__ bias,
                float* __restrict__ xt, float* __restrict__ agg)
{
    __shared__ float lx[16 * 132];               // 16x128 tile, stride 132 (bank-conflict free)
    const int tid  = threadIdx.x;
    const int wave = tid >> 5;
    const int lane = tid & 31;
    const int row0 = blockIdx.x * 16;

    // cooperative load of the 16x128 x-tile (float4, 2 per thread)
    {
        const float4* src = (const float4*)(X + (size_t)row0 * HH);
        for (int i = tid; i < 16 * 32; i += 256) {
            const int r  = i >> 5;
            const int c4 = i & 31;
            const float4 v = src[r * 32 + c4];
            float* d = &lx[r * 132 + c4 * 4];
            d[0] = v.x; d[1] = v.y; d[2] = v.z; d[3] = v.w;
        }
    }
    __syncthreads();

    const int n0   = wave * 16;          // output column base for this wave
    const int nl   = lane & 15;          // N within tile (B/C/D lane mapping)
    const int mrow = lane & 15;          // M within tile (A lane mapping)
    const int koff = (lane >> 4) << 1;   // lanes 16-31 hold K+2/K+3 (ISA 16x4 f32 layout)

    v8f c = {};
    for (int kk = 0; kk < HH; kk += 4) {
        v2f a, b;
        a.x = lx[mrow * 132 + kk + koff];
        a.y = lx[mrow * 132 + kk + koff + 1];
        b.x = W[(kk + koff)     * HH + n0 + nl];
        b.y = W[(kk + koff + 1) * HH + n0 + nl];
        c = __builtin_amdgcn_wmma_f32_16x16x4_f32(
                /*neg_a=*/false, a, /*neg_b=*/false, b,
                /*c_mod=*/(short)0, c, /*reuse_a=*/false, /*reuse_b=*/false);
    }

    // C/D layout: VGPR r -> M = r (lanes 0-15) or r+8 (lanes 16-31), N = lane%16
    const int   mbase = row0 + ((lane >> 4) << 3);
    const float bv    = bias[n0 + nl];
#pragma unroll
    for (int r = 0; r < 8; ++r) {
        const float  v   = c[r] + bv;
        const size_t idx = (size_t)(mbase + r) * HH + n0 + nl;
        xt[idx]  = v;
        agg[idx] = v;
    }
}

// ---------------------------------------------------------------------------
// Edge scatter: agg[dst] += xt[src]   (one wave per edge, float4 per lane)
// ---------------------------------------------------------------------------
__global__ void __launch_bounds__(256)
k_edge_scatter(const float* __restrict__ xt, const int* __restrict__ ei,
               float* __restrict__ agg)
{
    const int wid  = (blockIdx.x * 256 + threadIdx.x) >> 5;   // edge id
    const int lane = threadIdx.x & 31;
    if (wid >= EE) return;
    const int s = ei[wid];
    const int d = ei[EE + wid];
    const float4 v = ((const float4*)(xt + (size_t)s * HH))[lane];
    float* dp = agg + (size_t)d * HH + lane * 4;
    atomicAdd(dp + 0, v.x);
    atomicAdd(dp + 1, v.y);
    atomicAdd(dp + 2, v.z);
    atomicAdd(dp + 3, v.w);
}

// ---------------------------------------------------------------------------
// X = relu(LayerNorm(agg) * g + beta) + X     (one wave32 per row)
// ---------------------------------------------------------------------------
__global__ void __launch_bounds__(256)
k_ln_relu_res(const float* __restrict__ agg, const float* __restrict__ g,
              const float* __restrict__ beta, float* __restrict__ X)
{
    const int row  = (blockIdx.x * 256 + threadIdx.x) >> 5;
    const int lane = threadIdx.x & 31;
    if (row >= NN) return;
    const float4 v = ((const float4*)(agg + (size_t)row * HH))[lane];
    float s  = v.x + v.y + v.z + v.w;
    float sq = v.x * v.x + v.y * v.y + v.z * v.z + v.w * v.w;
#pragma unroll
    for (int off = 16; off >= 1; off >>= 1) {
        s  += __shfl_xor(s,  off, 32);
        sq += __shfl_xor(sq, off, 32);
    }
    const float mu  = s * (1.f / HH);
    const float var = sq * (1.f / HH) - mu * mu;
    const float inv = rsqrtf(var + LNEPS);
    const float4 gg = ((const float4*)g)[lane];
    const float4 bb = ((const float4*)beta)[lane];
    const float4 xo = ((const float4*)(X + (size_t)row * HH))[lane];
    float4 o;
    o.x = fmaxf((v.x - mu) * inv * gg.x + bb.x, 0.f) + xo.x;
    o.y = fmaxf((v.y - mu) * inv * gg.y + bb.y, 0.f) + xo.y;
    o.z = fmaxf((v.z - mu) * inv * gg.z + bb.z, 0.f) + xo.z;
    o.w = fmaxf((v.w - mu) * inv * gg.w + bb.w, 0.f) + xo.w;
    ((float4*)(X + (size_t)row * HH))[lane] = o;
}

// ---------------------------------------------------------------------------
// Column-sum reduction (for mean pooling)
// ---------------------------------------------------------------------------
__global__ void __launch_bounds__(128)
k_zero128(float* p) { p[threadIdx.x] = 0.f; }

__global__ void __launch_bounds__(128)
k_colsum(const float* __restrict__ X, float* __restrict__ colsum)
{
    const int j = threadIdx.x;
    float acc = 0.f;
    for (int r = blockIdx.x; r < NN; r += gridDim.x)
        acc += X[(size_t)r * HH + j];
    atomicAdd(&colsum[j], acc);
}

// emb = (colsum / N) @ Wp + bp
__global__ void __launch_bounds__(128)
k_pool_proj(const float* __restrict__ colsum, const float* __restrict__ Wp,
            const float* __restrict__ bp, float* __restrict__ emb)
{
    __shared__ float cm[HH];
    const int j = threadIdx.x;
    cm[j] = colsum[j] * (1.f / (float)NN);
    __syncthreads();
    float acc = bp[j];
    for (int i = 0; i < HH; ++i) acc = fmaf(cm[i], Wp[i * HH + j], acc);
    emb[j] = acc;
}

// ---------------------------------------------------------------------------
// Head MLP: combined[256] -> relu -> relu -> {rot[3], trans[3], sigmoid conf[1]}
// ---------------------------------------------------------------------------
__global__ void __launch_bounds__(128)
k_head(const float* __restrict__ rec_emb, const float* __restrict__ lig_emb,
       const float* __restrict__ W1, const float* __restrict__ b1,
       const float* __restrict__ W2, const float* __restrict__ b2,
       const float* __restrict__ Wr, const float* __restrict__ br,
       const float* __restrict__ Wt, const float* __restrict__ bt,
       const float* __restrict__ Wc, const float* __restrict__ bc,
       float* __restrict__ out)
{
    __shared__ float comb[2 * HH];
    __shared__ float f1[HH];
    __shared__ float f2[HH];
    const int j = threadIdx.x;
    comb[j]      = rec_emb[j];
    comb[HH + j] = lig_emb[j];
    __syncthreads();
    float acc = b1[j];
    for (int i = 0; i < 2 * HH; ++i) acc = fmaf(comb[i], W1[i * HH + j], acc);
    f1[j] = fmaxf(acc, 0.f);
    __syncthreads();
    acc = b2[j];
    for (int i = 0; i < HH; ++i) acc = fmaf(f1[i], W2[i * HH + j], acc);
    f2[j] = fmaxf(acc, 0.f);
    __syncthreads();
    if (j < 3) {
        float r = br[j], t = bt[j];
        for (int i = 0; i < HH; ++i) {
            r = fmaf(f2[i], Wr[i * 3 + j], r);
            t = fmaf(f2[i], Wt[i * 3 + j], t);
        }
        out[j]     = r;
        out[3 + j] = t;
    } else if (j == 3) {
        float cacc = bc[0];
        for (int i = 0; i < HH; ++i) cacc = fmaf(f2[i], Wc[i], cacc);
        out[6] = 1.f / (1.f + expf(-cacc));
    }
}

// ---------------------------------------------------------------------------
// d_in layout: top-level dict insertion order, params flattened as a JAX
// pytree (dict keys sorted at every level):
//  0 receptor_x  1 receptor_edge_index  2 ligand_x  3 ligand_edge_index
//  head  (sorted): 4 W1, 5 W2, 6 Wc, 7 Wr, 8 Wt, 9 b1, 10 b2, 11 bc, 12 br, 13 bt
//  lig   (sorted): 14 Wi, 15 Wp, 16 bi, 17 bp, 18.. layers[0..2]{W,b,beta,g}
//  rec   (sorted): 30 Wi, 31 Wp, 32 bi, 33 bp, 34.. layers[0..2]{W,b,beta,g}
// ---------------------------------------------------------------------------
extern "C" void kernel_launch(void* const* d_in, const int* in_sizes, int n_in,
                              void* d_out, int out_size, void* d_ws, size_t ws_size,
                              hipStream_t stream)
{
    (void)in_sizes; (void)n_in; (void)out_size; (void)ws_size;

    const float* rec_x  = (const float*)d_in[0];
    const int*   rec_ei = (const int*)  d_in[1];
    const float* lig_x  = (const float*)d_in[2];
    const int*   lig_ei = (const int*)  d_in[3];

    const float* hW1 = (const float*)d_in[4];
    const float* hW2 = (const float*)d_in[5];
    const float* hWc = (const float*)d_in[6];
    const float* hWr = (const float*)d_in[7];
    const float* hWt = (const float*)d_in[8];
    const float* hb1 = (const float*)d_in[9];
    const float* hb2 = (const float*)d_in[10];
    const float* hbc = (const float*)d_in[11];
    const float* hbr = (const float*)d_in[12];
    const float* hbt = (const float*)d_in[13];

    float* ws   = (float*)d_ws;
    float* X    = ws;
    float* XT   = ws + (size_t)NN * HH;
    float* AGG  = ws + 2 * (size_t)NN * HH;
    float* COLS = ws + 3 * (size_t)NN * HH;
    float* EREC = COLS + HH;
    float* ELIG = EREC + HH;

    auto run_encoder = [&](const float* xf, const int* ei, int pbase, float* emb) {
        const float* Wi = (const float*)d_in[pbase + 0];
        const float* Wp = (const float*)d_in[pbase + 1];
        const float* bi = (const float*)d_in[pbase + 2];
        const float* bp = (const float*)d_in[pbase + 3];

        k_input_proj<<<NN, 128, 0, stream>>>(xf, Wi, bi, X);

        for (int l = 0; l < 3; ++l) {
            const float* W    = (const float*)d_in[pbase + 4 + l * 4 + 0];
            const float* b    = (const float*)d_in[pbase + 4 + l * 4 + 1];
            const float* beta = (const float*)d_in[pbase + 4 + l * 4 + 2];
            const float* g    = (const float*)d_in[pbase + 4 + l * 4 + 3];
            k_gemm_wmma_f32<<<NN / 16, 256, 0, stream>>>(X, W, b, XT, AGG);
            k_edge_scatter<<<EE / 8, 256, 0, stream>>>(XT, ei, AGG);
            k_ln_relu_res<<<NN / 8, 256, 0, stream>>>(AGG, g, beta, X);
        }

        k_zero128<<<1, 128, 0, stream>>>(COLS);
        k_colsum<<<256, 128, 0, stream>>>(X, COLS);
        k_pool_proj<<<1, 128, 0, stream>>>(COLS, Wp, bp, emb);
    };

    run_encoder(rec_x, rec_ei, 30, EREC);   // receptor encoder params
    run_encoder(lig_x, lig_ei, 14, ELIG);   // ligand encoder params

    k_head<<<1, 128, 0, stream>>>(EREC, ELIG, hW1, hb1, hW2, hb2,
                                  hWr, hbr, hWt, hbt, hWc, hbc,
                                  (float*)d_out);
}